// Linear_MP_PDE_Solver_51840255262968
// MI455X (gfx1250) — compile-verified
//
#include <hip/hip_runtime.h>
#include <math.h>

#define NN   262144
#define EE   2097152
#define HF   128
#define TW   25
#define DT_C 0.01f
#define INV_L (1.0f/16.0f)
#define INV_T (1.0f/4.0f)

typedef __attribute__((ext_vector_type(2))) float v2f;
typedef __attribute__((ext_vector_type(8))) float v8f;

// swish via single-instruction v_rcp_f32 (avoid IEEE div expansion)
__device__ __forceinline__ float swishf(float x) {
    return x * __builtin_amdgcn_rcpf(1.0f + __expf(-x));
}

// ---------------------------------------------------------------- degree ----
__global__ __launch_bounds__(256) void deg_init_kernel(float* deg) {
    int i = blockIdx.x * 256 + threadIdx.x;
    deg[i] = 1.0f;                       // self-loop weight
}

__global__ __launch_bounds__(256) void deg_edge_kernel(const int* __restrict__ ei,
                                                       const float* __restrict__ pos,
                                                       float* deg) {
    int e = blockIdx.x * 256 + threadIdx.x;
    int s = ei[e];
    int d = ei[EE + e];
    float ew = fabsf(pos[2 * s + 1] - pos[2 * d + 1]) * INV_L;
    atomicAdd(&deg[d], ew);
}

__global__ __launch_bounds__(256) void deg_rsqrt_kernel(float* deg) {
    int i = blockIdx.x * 256 + threadIdx.x;
    deg[i] = __frsqrt_rn(deg[i]);        // deg >= 1 always (self loop)
}

// ------------------------------------------------------------- MLP (WMMA) ---
// One wave (32 threads) per 16-node tile.  Three fp32 WMMA GEMM stages:
//   h1 = swish(in(16x28,pad) @ W1(28x128) + b1)
//   h0 = swish(h1 @ W2 + b2)
//   xw = h0 @ gcnW
// writes:  xw,   hacc = h0 + DT*gcn_b + DT*dinv^2*xw   (self-loop folded in)
__global__ __launch_bounds__(32) void mlp_wmma_kernel(
        const float* __restrict__ u,   const float* __restrict__ pos,
        const float* __restrict__ W1,  const float* __restrict__ b1,
        const float* __restrict__ W2,  const float* __restrict__ b2,
        const float* __restrict__ gW,  const float* __restrict__ gb,
        const float* __restrict__ dinv,
        float* __restrict__ xw, float* __restrict__ hacc) {
    __shared__ float As[16 * 28];   // padded node_input tile (col 27 == 0)
    __shared__ float Hs[16 * HF];   // stage-1 output
    __shared__ float H2[16 * HF];   // stage-2 output (h0)

    const int base = blockIdx.x * 16;
    const int lane = threadIdx.x;           // 0..31, wave32
    const int l15  = lane & 15;
    const int half = lane >> 4;             // 0 | 1
    const int koff = half * 2;

    // build node_input tile: [u(25) | pos_x | pos_t | 0]
    for (int i = lane; i < 16 * 28; i += 32) {
        int r = i / 28, c = i % 28;
        int node = base + r;
        float v;
        if (c < TW)       v = u[node * TW + c];
        else if (c == 25) v = pos[2 * node + 1] * INV_L;
        else if (c == 26) v = pos[2 * node + 0] * INV_T;
        else              v = 0.0f;
        As[i] = v;
    }
    __syncthreads();

    // ---- stage 1: K = 28 (logical W1 row 27 == 0; branch-free: clamp the
    //      row index in-bounds and cndmask the value to zero) ----
    for (int n0 = 0; n0 < HF; n0 += 16) {
        v8f c;
        float bias = b1[n0 + l15];
        #pragma unroll
        for (int r = 0; r < 8; ++r) c[r] = bias;
        #pragma unroll
        for (int k0 = 0; k0 < 28; k0 += 4) {
            v2f a, b;
            a[0] = As[l15 * 28 + k0 + koff];
            a[1] = As[l15 * 28 + k0 + koff + 1];
            int kb0 = k0 + koff;            // <= 26 always
            int kb1 = kb0 + 1;              // <= 27
            int kc1 = kb1 > 26 ? 26 : kb1;  // clamped, always in-bounds
            float w0 = W1[kb0 * HF + n0 + l15];
            float w1 = W1[kc1 * HF + n0 + l15];
            b[0] = w0;
            b[1] = (kb1 < 27) ? w1 : 0.0f;  // v_cndmask, no branch
            c = __builtin_amdgcn_wmma_f32_16x16x4_f32(false, a, false, b,
                                                      (short)0, c, false, false);
        }
        #pragma unroll
        for (int r = 0; r < 8; ++r) {
            int row = r + half * 8;
            Hs[row * HF + n0 + l15] = swishf(c[r]);
        }
    }
    __syncthreads();

    // ---- stage 2: K = 128 ----
    for (int n0 = 0; n0 < HF; n0 += 16) {
        v8f c;
        float bias = b2[n0 + l15];
        #pragma unroll
        for (int r = 0; r < 8; ++r) c[r] = bias;
        for (int k0 = 0; k0 < HF; k0 += 4) {
            v2f a, b;
            a[0] = Hs[l15 * HF + k0 + koff];
            a[1] = Hs[l15 * HF + k0 + koff + 1];
            b[0] = W2[(k0 + koff)     * HF + n0 + l15];
            b[1] = W2[(k0 + koff + 1) * HF + n0 + l15];
            c = __builtin_amdgcn_wmma_f32_16x16x4_f32(false, a, false, b,
                                                      (short)0, c, false, false);
        }
        #pragma unroll
        for (int r = 0; r < 8; ++r) {
            int row = r + half * 8;
            H2[row * HF + n0 + l15] = swishf(c[r]);
        }
    }
    __syncthreads();

    // ---- stage 3: xw = h0 @ gcnW ; hacc = h0 + DT*(gcn_b + dinv^2 * xw) ----
    for (int n0 = 0; n0 < HF; n0 += 16) {
        v8f c;
        #pragma unroll
        for (int r = 0; r < 8; ++r) c[r] = 0.0f;
        for (int k0 = 0; k0 < HF; k0 += 4) {
            v2f a, b;
            a[0] = H2[l15 * HF + k0 + koff];
            a[1] = H2[l15 * HF + k0 + koff + 1];
            b[0] = gW[(k0 + koff)     * HF + n0 + l15];
            b[1] = gW[(k0 + koff + 1) * HF + n0 + l15];
            c = __builtin_amdgcn_wmma_f32_16x16x4_f32(false, a, false, b,
                                                      (short)0, c, false, false);
        }
        float gbv = gb[n0 + l15];
        #pragma unroll
        for (int r = 0; r < 8; ++r) {
            int row  = r + half * 8;
            int node = base + row;
            int col  = n0 + l15;
            float xv = c[r];
            float di = dinv[node];
            xw[(size_t)node * HF + col]   = xv;
            hacc[(size_t)node * HF + col] =
                H2[row * HF + col] + DT_C * (gbv + di * di * xv);
        }
    }
}

// -------------------------------------------------------------- scatter -----
// one wave per edge: hacc[dst] += DT * dinv[src]*|dx|/L*dinv[dst] * xw[src]
__global__ __launch_bounds__(256) void edge_scatter_kernel(
        const int* __restrict__ ei, const float* __restrict__ pos,
        const float* __restrict__ dinv, const float* __restrict__ xw,
        float* __restrict__ hacc) {
    int gid  = blockIdx.x * 256 + threadIdx.x;
    int edge = gid >> 5;
    int lane = threadIdx.x & 31;
    int s = ei[edge];
    int d = ei[EE + edge];
    float ew = fabsf(pos[2 * s + 1] - pos[2 * d + 1]) * INV_L;
    float sc = DT_C * dinv[s] * ew * dinv[d];
    const float4* xs = (const float4*)(xw + (size_t)s * HF);
    float4 v = xs[lane];
    float* hd = hacc + (size_t)d * HF + lane * 4;
    atomicAdd(hd + 0, sc * v.x);
    atomicAdd(hd + 1, sc * v.y);
    atomicAdd(hd + 2, sc * v.z);
    atomicAdd(hd + 3, sc * v.w);
}

// -------------------------------------------------- conv1 -> conv2 -> head --
// 64 threads/block, 1 node per thread; Hn staged through LDS (coalesced).
__global__ __launch_bounds__(64) void conv_head_kernel(
        const float* __restrict__ hacc,
        const float* __restrict__ cw1, const float* __restrict__ cb1,
        const float* __restrict__ cw2, const float* __restrict__ cb2,
        const float* __restrict__ lw,  const float* __restrict__ lb,
        float* __restrict__ out) {
    __shared__ float Hsh[64 * HF];          // 32 KB
    __shared__ float w1s[8 * 16], b1s[8];
    __shared__ float w2s[8 * 14];
    __shared__ float lws[TW * TW], lbs[TW];
    __shared__ float b2s;

    const int t    = threadIdx.x;
    const int base = blockIdx.x * 64;

    for (int i = t; i < 8 * 16;   i += 64) w1s[i] = cw1[i];
    for (int i = t; i < 8 * 14;   i += 64) w2s[i] = cw2[i];
    for (int i = t; i < TW * TW;  i += 64) lws[i] = lw[i];
    if (t < 8)  b1s[t] = cb1[t];
    if (t < TW) lbs[t] = lb[t];
    if (t == 0) b2s = cb2[0];
    for (int i = t; i < 64 * HF; i += 64)
        Hsh[i] = hacc[(size_t)base * HF + i];
    __syncthreads();

    const float* H = &Hsh[t * HF];
    float acc2[TW];
    #pragma unroll
    for (int q = 0; q < TW; ++q) acc2[q] = b2s;

    for (int c = 0; c < 8; ++c) {
        float o1[38];
        float bc = b1s[c];
        #pragma unroll
        for (int p = 0; p < 38; ++p) {
            float a = bc;
            #pragma unroll
            for (int k = 0; k < 16; ++k) a += H[p * 3 + k] * w1s[c * 16 + k];
            o1[p] = swishf(a);
        }
        #pragma unroll
        for (int q = 0; q < TW; ++q) {
            float s = 0.0f;
            #pragma unroll
            for (int k = 0; k < 14; ++k) s += o1[q + k] * w2s[c * 14 + k];
            acc2[q] += s;
        }
    }
    #pragma unroll
    for (int q = 0; q < TW; ++q) acc2[q] = swishf(acc2[q]);

    float* op = out + (size_t)(base + t) * TW;
    #pragma unroll
    for (int k = 0; k < TW; ++k) {
        float a = lbs[k];
        #pragma unroll
        for (int q = 0; q < TW; ++q) a += acc2[q] * lws[k * TW + q];
        op[k] = a;
    }
}

// ----------------------------------------------------------------- launch ---
extern "C" void kernel_launch(void* const* d_in, const int* in_sizes, int n_in,
                              void* d_out, int out_size, void* d_ws, size_t ws_size,
                              hipStream_t stream) {
    const float* u   = (const float*)d_in[0];
    const float* pos = (const float*)d_in[1];
    const int*   ei  = (const int*)  d_in[2];
    const float* W1  = (const float*)d_in[3];
    const float* b1  = (const float*)d_in[4];
    const float* W2  = (const float*)d_in[5];
    const float* b2  = (const float*)d_in[6];
    const float* gW  = (const float*)d_in[7];
    const float* gb  = (const float*)d_in[8];
    const float* cw1 = (const float*)d_in[9];
    const float* cb1 = (const float*)d_in[10];
    const float* cw2 = (const float*)d_in[11];
    const float* cb2 = (const float*)d_in[12];
    const float* lw  = (const float*)d_in[13];
    const float* lb  = (const float*)d_in[14];
    float* out = (float*)d_out;

    float* xw   = (float*)d_ws;                     // N*128 floats
    float* hacc = xw   + (size_t)NN * HF;           // N*128 floats
    float* dinv = hacc + (size_t)NN * HF;           // N floats

    deg_init_kernel <<<NN / 256, 256, 0, stream>>>(dinv);
    deg_edge_kernel <<<EE / 256, 256, 0, stream>>>(ei, pos, dinv);
    deg_rsqrt_kernel<<<NN / 256, 256, 0, stream>>>(dinv);
    mlp_wmma_kernel <<<NN / 16,   32, 0, stream>>>(u, pos, W1, b1, W2, b2,
                                                   gW, gb, dinv, xw, hacc);
    edge_scatter_kernel<<<EE / 8, 256, 0, stream>>>(ei, pos, dinv, xw, hacc);
    conv_head_kernel<<<NN / 64,   64, 0, stream>>>(hacc, cw1, cb1, cw2, cb2,
                                                   lw, lb, out);
}